// MultiHeadAttention_58351425683782
// MI455X (gfx1250) — compile-verified
//
#include <hip/hip_runtime.h>
#include <math.h>

// Problem constants (from reference)
#define B_  4
#define S_  2048
#define D_  1024
#define H_  16
#define DK_ 64

typedef float v2f __attribute__((ext_vector_type(2)));
typedef float v8f __attribute__((ext_vector_type(8)));
typedef int   v4i __attribute__((ext_vector_type(4)));

// D(16x16,f32) = A(16x4,f32) * B(4x16,f32) + C  — dense fp32 WMMA on gfx1250.
// Lane layout (wave32): g = lane>>4, t = lane&15
//   A: a.{x,y} = A[M=t][K=2g+{0,1}]
//   B: b.{x,y} = B[K=2g+{0,1}][N=t]
//   C/D: c[r]  = C[M=8g+r][N=t]
__device__ __forceinline__ v8f wmma_f32(v2f a, v2f b, v8f c) {
  return __builtin_amdgcn_wmma_f32_16x16x4_f32(false, a, false, b, (short)0, c,
                                               false, false);
}

// ---- CDNA5 async global->LDS copy (ASYNCcnt-tracked), with safe fallback ----
#if defined(__gfx1250__) && __has_builtin(__builtin_amdgcn_global_load_async_to_lds_b128)
#define USE_ASYNC_LDS 1
#else
#define USE_ASYNC_LDS 0
#endif

__device__ __forceinline__ void cp16(void* lds, const void* gmem) {
#if USE_ASYNC_LDS
  // Real signature (from compiler diagnostic): param0 = v4i in addrspace(1).
  __builtin_amdgcn_global_load_async_to_lds_b128(
      (__attribute__((address_space(1))) v4i*)gmem,
      (__attribute__((address_space(3))) v4i*)lds, /*offset=*/0, /*cpol=*/0);
#else
  *(float4*)lds = *(const float4*)gmem;
#endif
}
__device__ __forceinline__ void cp_wait() {
#if USE_ASYNC_LDS
  asm volatile("s_wait_asynccnt 0" ::: "memory");
#endif
}

// ---------------------------------------------------------------------------
// Kernel 1: per-head projections  Kp/Vp[hb][s][dk] = sum_d value[b][s][d]*W[h][d][dk]
// grid = (S/128, H*B, 2), block = 256. 8 waves = 4x2, each wave owns a 32x32
// tile (2x2 WMMA accumulators => 1 LDS fragment load per WMMA). KC=32,
// double-buffered async global->LDS staging.
// ---------------------------------------------------------------------------
__global__ __launch_bounds__(256)
void proj_kernel(const float* __restrict__ value, const float* __restrict__ Wk,
                 const float* __restrict__ Wv, float* __restrict__ Kp,
                 float* __restrict__ Vp) {
  __shared__ __align__(16) float A2[2][128][36];   // value tile, stride 36 (bank 4*(9t%16))
  __shared__ __align__(16) float Bt2[2][64][36];   // weight tile TRANSPOSED [dk][d]

  const int tid  = threadIdx.x;
  const int lane = tid & 31, wave = tid >> 5;
  const int g = lane >> 4, t = lane & 15;
  const int wr = wave >> 1;            // 0..3 : 32-row band
  const int wc = wave & 1;             // 0..1 : 32-col band
  const int row0 = blockIdx.x * 128;
  const int hb   = blockIdx.y;         // h*B + b
  const int h    = hb / B_;
  const float* W   = (blockIdx.z == 0) ? Wk : Wv;
  float*       Out = (blockIdx.z == 0) ? Kp : Vp;
  const float* vbase = value + (size_t)(hb % B_) * S_ * D_;
  const float* wbase = W + (size_t)h * D_ * DK_;

  v8f zero = {0.f, 0.f, 0.f, 0.f, 0.f, 0.f, 0.f, 0.f};
  v8f acc00 = zero, acc01 = zero, acc10 = zero, acc11 = zero;

  // A chunk: 128 rows x 32 k (async, contiguous 16B pieces)
  auto loadA = [&](int buf, int d0) {
#pragma unroll
    for (int i = 0; i < 4; ++i) {
      int q = tid + 256 * i;                  // 1024 x 16B chunks
      int r = q >> 3, cc = (q & 7) * 4;
      cp16(&A2[buf][r][cc], vbase + (size_t)(row0 + r) * D_ + d0 + cc);
    }
  };
  // B chunk: W[d0..d0+31][0..63] -> transposed Bt[dk][d] (sync scatter)
  auto loadB = [&](int buf, int d0) {
#pragma unroll
    for (int i = 0; i < 2; ++i) {
      int q = tid + 256 * i;                  // 512 x float4 (coalesced reads)
      int r = q >> 4, cc = (q & 15) * 4;
      float4 w = *(const float4*)(wbase + (size_t)(d0 + r) * DK_ + cc);
      Bt2[buf][cc + 0][r] = w.x;
      Bt2[buf][cc + 1][r] = w.y;
      Bt2[buf][cc + 2][r] = w.z;
      Bt2[buf][cc + 3][r] = w.w;
    }
  };

  loadA(0, 0);
  loadB(0, 0);
  cp_wait();
  __syncthreads();

  const int NC = D_ / 32;
  for (int ch = 0; ch < NC; ++ch) {
    const int cur = ch & 1;
    if (ch + 1 < NC) {               // prefetch next chunk into other buffer
      loadA(cur ^ 1, (ch + 1) * 32);
      loadB(cur ^ 1, (ch + 1) * 32);
    }
#pragma unroll
    for (int kk = 0; kk < 32; kk += 4) {
      const int k2 = kk + 2 * g;
      v2f a0, a1, b0, b1;
      a0.x = A2[cur][wr * 32 + t][k2];        a0.y = A2[cur][wr * 32 + t][k2 + 1];
      a1.x = A2[cur][wr * 32 + 16 + t][k2];   a1.y = A2[cur][wr * 32 + 16 + t][k2 + 1];
      b0.x = Bt2[cur][wc * 32 + t][k2];       b0.y = Bt2[cur][wc * 32 + t][k2 + 1];
      b1.x = Bt2[cur][wc * 32 + 16 + t][k2];  b1.y = Bt2[cur][wc * 32 + 16 + t][k2 + 1];
      acc00 = wmma_f32(a0, b0, acc00);
      acc01 = wmma_f32(a0, b1, acc01);
      acc10 = wmma_f32(a1, b0, acc10);
      acc11 = wmma_f32(a1, b1, acc11);
    }
    cp_wait();
    __syncthreads();
  }

#pragma unroll
  for (int r = 0; r < 8; ++r) {
    int rowa = row0 + wr * 32 + 8 * g + r;
    int rowb = rowa + 16;
    size_t basea = ((size_t)hb * S_ + rowa) * DK_;
    size_t baseb = ((size_t)hb * S_ + rowb) * DK_;
    Out[basea + wc * 32 + t]      = acc00[r];
    Out[basea + wc * 32 + 16 + t] = acc01[r];
    Out[baseb + wc * 32 + t]      = acc10[r];
    Out[baseb + wc * 32 + 16 + t] = acc11[r];
  }
}

// ---------------------------------------------------------------------------
// Kernel 2: flash attention (faithful quirks: scores = Vp @ Kp^T, ctx = P @ Kp)
// grid = (S/128, H*B), block = 256; each wave owns 16 query rows, so all
// softmax row statistics stay inside one wave32 (shfl over 16-lane halves).
// Kj staged via async global->LDS; padding 68 => conflict-free transposed reads.
// ---------------------------------------------------------------------------
__global__ __launch_bounds__(256)
void attn_kernel(const float* __restrict__ Kp, const float* __restrict__ Vp,
                 const unsigned char* __restrict__ mask,
                 float* __restrict__ concat) {
  __shared__ __align__(16) float Kj[64][68];     // key block tile (row stride 272B)
  __shared__ __align__(16) float P[8][16][68];   // per-wave probability strip

  const int tid  = threadIdx.x;
  const int lane = tid & 31, wave = tid >> 5;
  const int g = lane >> 4, t = lane & 15;
  const int hb = blockIdx.y;
  const int h = hb / B_, b = hb % B_;
  const int srow0 = blockIdx.x * 128 + wave * 16;
  const float scale = 0.125f;  // DK^-0.5

  // Cache this wave's 16 query rows (V-projection) as WMMA A-fragments.
  v2f a_q[16];
  {
    const float* vp = Vp + ((size_t)hb * S_ + srow0 + t) * DK_;
#pragma unroll
    for (int ks = 0; ks < 16; ++ks) {
      a_q[ks].x = vp[4 * ks + 2 * g];
      a_q[ks].y = vp[4 * ks + 2 * g + 1];
    }
  }

  v8f zero = {0.f, 0.f, 0.f, 0.f, 0.f, 0.f, 0.f, 0.f};
  v8f o[4];
#pragma unroll
  for (int n = 0; n < 4; ++n) o[n] = zero;
  float m_i[8], l_i[8];
#pragma unroll
  for (int r = 0; r < 8; ++r) { m_i[r] = -INFINITY; l_i[r] = 0.f; }

  const unsigned char* mbase = mask + (size_t)b * S_ * S_;

  for (int j0 = 0; j0 < S_; j0 += 64) {
    __syncthreads();  // all waves done reading previous Kj
#pragma unroll
    for (int i = 0; i < 4; ++i) {
      int q = tid + 256 * i;                  // 64 rows x 16 chunks of 16B
      int r = q >> 4, cc = (q & 15) * 4;
      cp16(&Kj[r][cc], Kp + ((size_t)hb * S_ + j0 + r) * DK_ + cc);
    }
    cp_wait();
    __syncthreads();

    // scores tile: 16 rows x 64 key-cols, via 4 WMMA N-tiles (B = Kp^T)
    v8f sc[4];
#pragma unroll
    for (int n = 0; n < 4; ++n) {
      v8f c = zero;
#pragma unroll
      for (int ks = 0; ks < 16; ++ks) {
        v2f bfrag;  // transposed read: lane strides rows (bank 4t, conflict-free)
        bfrag.x = Kj[n * 16 + t][4 * ks + 2 * g];
        bfrag.y = Kj[n * 16 + t][4 * ks + 2 * g + 1];
        c = wmma_f32(a_q[ks], bfrag, c);
      }
      sc[n] = c;
    }

    // scale + mask (mask==true -> -inf), faithful to reference
#pragma unroll
    for (int n = 0; n < 4; ++n) {
#pragma unroll
      for (int r = 0; r < 8; ++r) {
        float v = sc[n][r] * scale;
        int qrow = srow0 + 8 * g + r;
        int kcol = j0 + n * 16 + t;
        if (mbase[(size_t)qrow * S_ + kcol]) v = -INFINITY;
        sc[n][r] = v;
      }
    }

    // online-softmax row stats (rows live entirely inside this wave)
    float mnew[8], alpha[8];
#pragma unroll
    for (int r = 0; r < 8; ++r) {
      float mx = fmaxf(fmaxf(sc[0][r], sc[1][r]), fmaxf(sc[2][r], sc[3][r]));
#pragma unroll
      for (int off = 1; off < 16; off <<= 1)
        mx = fmaxf(mx, __shfl_xor(mx, off, 32));
      mnew[r]  = fmaxf(m_i[r], mx);
      alpha[r] = __expf(m_i[r] - mnew[r]);
      m_i[r]   = mnew[r];
    }
#pragma unroll
    for (int r = 0; r < 8; ++r) {
      float s = 0.f;
#pragma unroll
      for (int n = 0; n < 4; ++n) {
        float p = __expf(sc[n][r] - mnew[r]);
        sc[n][r] = p;
        s += p;
      }
#pragma unroll
      for (int off = 1; off < 16; off <<= 1) s += __shfl_xor(s, off, 32);
      l_i[r] = l_i[r] * alpha[r] + s;
    }

    // stage P into this wave's private LDS strip; wave-local DS fence only
#pragma unroll
    for (int n = 0; n < 4; ++n)
#pragma unroll
      for (int r = 0; r < 8; ++r) P[wave][8 * g + r][n * 16 + t] = sc[n][r];
    asm volatile("s_wait_dscnt 0" ::: "memory");

    // reload P as A-fragments (lane strides rows: stride 68 -> bank 4t)
    v2f a_p[16];
#pragma unroll
    for (int ks = 0; ks < 16; ++ks) {
      a_p[ks].x = P[wave][t][4 * ks + 2 * g];
      a_p[ks].y = P[wave][t][4 * ks + 2 * g + 1];
    }

    // ctx += P @ Kj (faithful quirk: uses K-projection as "values")
#pragma unroll
    for (int n = 0; n < 4; ++n) {
      v8f c;
#pragma unroll
      for (int r = 0; r < 8; ++r) c[r] = o[n][r] * alpha[r];
#pragma unroll
      for (int ks = 0; ks < 16; ++ks) {
        v2f bfrag;
        bfrag.x = Kj[4 * ks + 2 * g][n * 16 + t];
        bfrag.y = Kj[4 * ks + 2 * g + 1][n * 16 + t];
        c = wmma_f32(a_p[ks], bfrag, c);
      }
      o[n] = c;
    }
  }

  // normalize and store straight into head-concat layout (B, S, H*DK)
#pragma unroll
  for (int n = 0; n < 4; ++n)
#pragma unroll
    for (int r = 0; r < 8; ++r) {
      int s  = srow0 + 8 * g + r;
      int dk = n * 16 + t;
      concat[((size_t)(b * S_ + s) * H_ + h) * DK_ + dk] = o[n][r] / l_i[r];
    }
}

// ---------------------------------------------------------------------------
// Kernel 3: out = concat @ Wf^T + bf     (M=B*S, N=D, K=H*DK)
// grid = (B*S/128, D/64), block = 256. Same 32x32-per-wave register blocking
// and async double-buffered staging as kernel 1. Wf tile stored [n][k] and
// read transposed (naturally contiguous in global on the k axis).
// ---------------------------------------------------------------------------
__global__ __launch_bounds__(256)
void final_kernel(const float* __restrict__ concat, const float* __restrict__ Wf,
                  const float* __restrict__ bf, float* __restrict__ out) {
  __shared__ __align__(16) float A2[2][128][36];
  __shared__ __align__(16) float W2[2][64][36];   // [output col n][k]

  const int tid  = threadIdx.x;
  const int lane = tid & 31, wave = tid >> 5;
  const int g = lane >> 4, t = lane & 15;
  const int wr = wave >> 1;            // 0..3
  const int wc = wave & 1;             // 0..1
  const int row0 = blockIdx.x * 128;   // over B*S
  const int col0 = blockIdx.y * 64;    // over D

  v8f zero = {0.f, 0.f, 0.f, 0.f, 0.f, 0.f, 0.f, 0.f};
  v8f acc00 = zero, acc01 = zero, acc10 = zero, acc11 = zero;

  auto loadA = [&](int buf, int k0) {
#pragma unroll
    for (int i = 0; i < 4; ++i) {
      int q = tid + 256 * i;
      int r = q >> 3, cc = (q & 7) * 4;
      cp16(&A2[buf][r][cc], concat + (size_t)(row0 + r) * (H_ * DK_) + k0 + cc);
    }
  };
  auto loadW = [&](int buf, int k0) {
#pragma unroll
    for (int i = 0; i < 2; ++i) {
      int q = tid + 256 * i;                  // 64 rows x 8 chunks
      int r = q >> 3, cc = (q & 7) * 4;
      cp16(&W2[buf][r][cc], Wf + (size_t)(col0 + r) * (H_ * DK_) + k0 + cc);
    }
  };

  loadA(0, 0);
  loadW(0, 0);
  cp_wait();
  __syncthreads();

  const int NC = (H_ * DK_) / 32;
  for (int ch = 0; ch < NC; ++ch) {
    const int cur = ch & 1;
    if (ch + 1 < NC) {
      loadA(cur ^ 1, (ch + 1) * 32);
      loadW(cur ^ 1, (ch + 1) * 32);
    }
#pragma unroll
    for (int kk = 0; kk < 32; kk += 4) {
      const int k2 = kk + 2 * g;
      v2f a0, a1, b0, b1;
      a0.x = A2[cur][wr * 32 + t][k2];        a0.y = A2[cur][wr * 32 + t][k2 + 1];
      a1.x = A2[cur][wr * 32 + 16 + t][k2];   a1.y = A2[cur][wr * 32 + 16 + t][k2 + 1];
      b0.x = W2[cur][wc * 32 + t][k2];        b0.y = W2[cur][wc * 32 + t][k2 + 1];
      b1.x = W2[cur][wc * 32 + 16 + t][k2];   b1.y = W2[cur][wc * 32 + 16 + t][k2 + 1];
      acc00 = wmma_f32(a0, b0, acc00);
      acc01 = wmma_f32(a0, b1, acc01);
      acc10 = wmma_f32(a1, b0, acc10);
      acc11 = wmma_f32(a1, b1, acc11);
    }
    cp_wait();
    __syncthreads();
  }

  const float bias0 = bf[col0 + wc * 32 + t];
  const float bias1 = bf[col0 + wc * 32 + 16 + t];
#pragma unroll
  for (int r = 0; r < 8; ++r) {
    int rowa = row0 + wr * 32 + 8 * g + r;
    int rowb = rowa + 16;
    out[(size_t)rowa * D_ + col0 + wc * 32 + t]      = acc00[r] + bias0;
    out[(size_t)rowa * D_ + col0 + wc * 32 + 16 + t] = acc01[r] + bias1;
    out[(size_t)rowb * D_ + col0 + wc * 32 + t]      = acc10[r] + bias0;
    out[(size_t)rowb * D_ + col0 + wc * 32 + 16 + t] = acc11[r] + bias1;
  }
}

// ---------------------------------------------------------------------------
extern "C" void kernel_launch(void* const* d_in, const int* in_sizes, int n_in,
                              void* d_out, int out_size, void* d_ws,
                              size_t ws_size, hipStream_t stream) {
  (void)in_sizes; (void)n_in; (void)out_size; (void)ws_size;
  const float* value        = (const float*)d_in[0];
  const unsigned char* mask = (const unsigned char*)d_in[1];
  // d_in[2] = Wq : computed-but-unused in the reference -> skipped entirely
  const float* Wk = (const float*)d_in[3];
  const float* Wv = (const float*)d_in[4];
  const float* Wf = (const float*)d_in[5];
  const float* bf = (const float*)d_in[6];
  float* out = (float*)d_out;

  const size_t nProj = (size_t)H_ * B_ * S_ * DK_;  // 8.39M floats each
  float* Kp     = (float*)d_ws;
  float* Vp     = Kp + nProj;
  float* concat = Vp + nProj;

  dim3 blk(256);
  proj_kernel<<<dim3(S_ / 128, H_ * B_, 2), blk, 0, stream>>>(value, Wk, Wv, Kp, Vp);
  attn_kernel<<<dim3(S_ / 128, H_ * B_), blk, 0, stream>>>(Kp, Vp, mask, concat);
  final_kernel<<<dim3((B_ * S_) / 128, D_ / 64), blk, 0, stream>>>(concat, Wf, bf, out);
}